// ANModel_35485019800323
// MI455X (gfx1250) — compile-verified
//
#include <hip/hip_runtime.h>

typedef __attribute__((ext_vector_type(2))) float v2f;
typedef __attribute__((ext_vector_type(8))) float v8f;

#define NB   512
#define BATCH 8
#define EDIM 8
#define DD   16
#define MM   6
#define HH   64
#define SRCF 64
#define OBSF 32

// workspace layout (float offsets)
#define WS_X      0          // 4096*16
#define WS_XAGG   65536      // 4096*16
#define WS_EAGG   131072     // 4096*8
#define WS_DEG    163840     // 512
#define WS_WC     164352     // 128  (we1 @ W3)
#define WS_BEW3   164480     // 16   (be1 @ W3)
#define WS_XAD    164608     // 4096*16  x_adapt
// total = 230144 floats = 920,576 bytes

static __device__ __forceinline__ v8f wmma_f32(v2f a, v2f b, v8f c) {
  return __builtin_amdgcn_wmma_f32_16x16x4_f32(false, a, false, b, (short)0, c,
                                               false, false);
}

// ---------------------------------------------------------------- k_setup
// Wc = we1 @ W3  (8x16),  beW3 = be1 @ W3  (16)
__global__ void __launch_bounds__(128) k_setup(const float* __restrict__ ge_we,
                                               const float* __restrict__ ge_be,
                                               const float* __restrict__ ge_wemb,
                                               float* __restrict__ ws) {
  const float* we1 = ge_we + EDIM * DD;              // layer 1
  const float* be1 = ge_be + DD;
  const float* W3  = ge_wemb + 3 * DD * DD + 2 * DD * DD;  // wemb1[32:48]
  int t = threadIdx.x;
  if (t < EDIM * DD) {
    int e = t >> 4, k = t & 15;
    float s = 0.f;
    for (int d = 0; d < DD; ++d) s += we1[e * DD + d] * W3[d * DD + k];
    ws[WS_WC + t] = s;
  }
  if (t < DD) {
    float s = 0.f;
    for (int d = 0; d < DD; ++d) s += be1[d] * W3[d * DD + t];
    ws[WS_BEW3 + t] = s;
  }
}

// ---------------------------------------------------------------- k_embed
// x = obs_all @ w_obs + b_obs  (4096x32 @ 32x16), WMMA f32 16x16x4.
// One wave per 16-row tile; 8 k-steps.
__global__ void __launch_bounds__(256) k_embed(const float* __restrict__ obs_all,
                                               const float* __restrict__ w_obs,
                                               const float* __restrict__ b_obs,
                                               float* __restrict__ ws) {
  int lane = threadIdx.x & 31;
  int tile = blockIdx.x * 8 + (threadIdx.x >> 5);  // 0..255
  int q0 = tile << 4;
  int n = lane & 15;
  int hi = lane >> 4;
  int kb = hi << 1;   // K offset for upper half-wave
  float* xw = ws + WS_X;

  v8f c;
  float bo = b_obs[n];
#pragma unroll
  for (int r = 0; r < 8; ++r) c[r] = bo;

  const float* arow = obs_all + (size_t)(q0 + n) * OBSF;
#pragma unroll
  for (int k0 = 0; k0 < OBSF; k0 += 4) {
    v2f a = *reinterpret_cast<const v2f*>(arow + k0 + kb);
    v2f b;
    b.x = w_obs[(k0 + kb) * DD + n];
    b.y = w_obs[(k0 + kb + 1) * DD + n];
    c = wmma_f32(a, b, c);
  }
  int m0 = q0 + (hi << 3);
#pragma unroll
  for (int r = 0; r < 8; ++r) xw[(size_t)(m0 + r) * DD + n] = c[r];
}

// ------------------------------------------------------------ k_aggregate
// Per (b,i): eagg = sum_j adj[i,j]*EA[b,i,j,:], xagg = sum_j adj[i,j]*x[b,j,:],
// deg[i] = sum_j adj[i,j].  One wave per (b,i); the 67 MB stream.
__global__ void __launch_bounds__(256) k_aggregate(const float* __restrict__ ea,
                                                   const float* __restrict__ A,
                                                   const float* __restrict__ PA,
                                                   float* __restrict__ ws) {
  int lane = threadIdx.x & 31;
  int wid = blockIdx.x * 8 + (threadIdx.x >> 5);  // 0..4095 = b*512+i
  int b = wid >> 9;
  int i = wid & (NB - 1);

  const float* xw = ws + WS_X + (size_t)b * NB * DD;
  const float* earow = ea + (size_t)wid * NB * EDIM;
  const float* Ar = A + (size_t)i * NB;
  const float* Pr = PA + (size_t)i * NB;

  float eacc[EDIM], xacc[DD], sw = 0.f;
#pragma unroll
  for (int e = 0; e < EDIM; ++e) eacc[e] = 0.f;
#pragma unroll
  for (int k = 0; k < DD; ++k) xacc[k] = 0.f;

  for (int it = 0; it < NB / 32; ++it) {
    int j = lane + (it << 5);
    __builtin_prefetch(earow + (size_t)(j + 32) * EDIM, 0, 1);
    float w = Ar[j] + Pr[j];
    sw += w;
    const float4* ep = reinterpret_cast<const float4*>(earow + (size_t)j * EDIM);
    float4 e0 = ep[0], e1 = ep[1];
    eacc[0] += w * e0.x; eacc[1] += w * e0.y; eacc[2] += w * e0.z; eacc[3] += w * e0.w;
    eacc[4] += w * e1.x; eacc[5] += w * e1.y; eacc[6] += w * e1.z; eacc[7] += w * e1.w;
    const float4* xp = reinterpret_cast<const float4*>(xw + (size_t)j * DD);
    float4 x0 = xp[0], x1 = xp[1], x2 = xp[2], x3 = xp[3];
    xacc[0]  += w * x0.x; xacc[1]  += w * x0.y; xacc[2]  += w * x0.z; xacc[3]  += w * x0.w;
    xacc[4]  += w * x1.x; xacc[5]  += w * x1.y; xacc[6]  += w * x1.z; xacc[7]  += w * x1.w;
    xacc[8]  += w * x2.x; xacc[9]  += w * x2.y; xacc[10] += w * x2.z; xacc[11] += w * x2.w;
    xacc[12] += w * x3.x; xacc[13] += w * x3.y; xacc[14] += w * x3.z; xacc[15] += w * x3.w;
  }

#pragma unroll
  for (int off = 16; off > 0; off >>= 1) {
    sw += __shfl_xor(sw, off, 32);
#pragma unroll
    for (int e = 0; e < EDIM; ++e) eacc[e] += __shfl_xor(eacc[e], off, 32);
#pragma unroll
    for (int k = 0; k < DD; ++k) xacc[k] += __shfl_xor(xacc[k], off, 32);
  }

  if (lane == 0) {
    float4* eo = reinterpret_cast<float4*>(ws + WS_EAGG + (size_t)wid * EDIM);
    eo[0] = make_float4(eacc[0], eacc[1], eacc[2], eacc[3]);
    eo[1] = make_float4(eacc[4], eacc[5], eacc[6], eacc[7]);
    float4* xo = reinterpret_cast<float4*>(ws + WS_XAGG + (size_t)wid * DD);
    xo[0] = make_float4(xacc[0], xacc[1], xacc[2], xacc[3]);
    xo[1] = make_float4(xacc[4], xacc[5], xacc[6], xacc[7]);
    xo[2] = make_float4(xacc[8], xacc[9], xacc[10], xacc[11]);
    xo[3] = make_float4(xacc[12], xacc[13], xacc[14], xacc[15]);
    if (b == 0) ws[WS_DEG + i] = sw;
  }
}

// -------------------------------------------------------------- k_combine
// x_adapt = relu( x@R + (deg*x)@W1 + xagg@W2 + eagg@Wc
//                 + res_b1 + deg*(bemb1 + beW3) )
// One wave per 16-row tile; 14 chained WMMAs; bias folded into C init.
__global__ void __launch_bounds__(256) k_combine(const float* __restrict__ res_w,
                                                 const float* __restrict__ res_b,
                                                 const float* __restrict__ ge_wemb,
                                                 const float* __restrict__ ge_bemb,
                                                 float* __restrict__ ws) {
  int lane = threadIdx.x & 31;
  int tile = blockIdx.x * 8 + (threadIdx.x >> 5);  // 0..255
  int i0 = (tile & 31) << 4;
  int q0 = tile << 4;
  int n = lane & 15, hi = lane >> 4;
  int kb = hi << 1, mb = hi << 3;

  const float* R  = res_w + DD * DD;        // layer 1
  const float* W1 = ge_wemb + 3 * DD * DD;  // wemb1[0:16]
  const float* W2 = W1 + DD * DD;           // wemb1[16:32]
  const float* Wc = ws + WS_WC;
  const float* xw = ws + WS_X;
  const float* xg = ws + WS_XAGG;
  const float* eg = ws + WS_EAGG;
  const float* dg = ws + WS_DEG;

  float rb  = res_b[DD + n];
  float bb2 = ge_bemb[DD + n] + ws[WS_BEW3 + n];

  v8f c;
#pragma unroll
  for (int r = 0; r < 8; ++r) c[r] = rb + dg[i0 + mb + r] * bb2;

  float dega = dg[i0 + n];
  const float* xr  = xw + (size_t)(q0 + n) * DD;
  const float* xgr = xg + (size_t)(q0 + n) * DD;
  const float* egr = eg + (size_t)(q0 + n) * EDIM;

#pragma unroll
  for (int k0 = 0; k0 < DD; k0 += 4) {  // x @ R
    v2f a = *reinterpret_cast<const v2f*>(xr + k0 + kb);
    v2f b; b.x = R[(k0 + kb) * DD + n]; b.y = R[(k0 + kb + 1) * DD + n];
    c = wmma_f32(a, b, c);
  }
#pragma unroll
  for (int k0 = 0; k0 < DD; k0 += 4) {  // (deg*x) @ W1
    v2f a = *reinterpret_cast<const v2f*>(xr + k0 + kb);
    a.x *= dega; a.y *= dega;
    v2f b; b.x = W1[(k0 + kb) * DD + n]; b.y = W1[(k0 + kb + 1) * DD + n];
    c = wmma_f32(a, b, c);
  }
#pragma unroll
  for (int k0 = 0; k0 < DD; k0 += 4) {  // xagg @ W2
    v2f a = *reinterpret_cast<const v2f*>(xgr + k0 + kb);
    v2f b; b.x = W2[(k0 + kb) * DD + n]; b.y = W2[(k0 + kb + 1) * DD + n];
    c = wmma_f32(a, b, c);
  }
#pragma unroll
  for (int k0 = 0; k0 < EDIM; k0 += 4) {  // eagg @ Wc
    v2f a = *reinterpret_cast<const v2f*>(egr + k0 + kb);
    v2f b; b.x = Wc[(k0 + kb) * DD + n]; b.y = Wc[(k0 + kb + 1) * DD + n];
    c = wmma_f32(a, b, c);
  }

  float* xa = ws + WS_XAD;
#pragma unroll
  for (int r = 0; r < 8; ++r)
    xa[(size_t)(q0 + mb + r) * DD + n] = fmaxf(c[r], 0.f);
}

// ----------------------------------------------------------------- k_head
// Gather + gated MLP head; one block per batch row.
__global__ void __launch_bounds__(128) k_head(const float* __restrict__ obs,
                                              const int* __restrict__ ridxs,
                                              const int* __restrict__ rni,
                                              const int* __restrict__ rnm,
                                              const float* __restrict__ w_src,
                                              const float* __restrict__ b_src,
                                              const float* __restrict__ gated_w,
                                              const float* __restrict__ gated_b,
                                              const float* __restrict__ w1,
                                              const float* __restrict__ b1,
                                              const float* __restrict__ w2,
                                              const float* __restrict__ b2,
                                              const float* __restrict__ aw,
                                              const float* __restrict__ ab,
                                              const float* __restrict__ ws,
                                              float* __restrict__ out) {
  __shared__ float s_in[192];
  __shared__ float s_g[32];
  __shared__ float s_h[16];
  __shared__ float s_h1[64];
  __shared__ float s_h2[64];
  int b = blockIdx.x;
  int t = threadIdx.x;
  const float* xa = ws + WS_XAD + (size_t)b * NB * DD;

  if (t < 96) {
    float s = b_src[t];
    for (int d = 0; d < SRCF; ++d) s += obs[b * SRCF + d] * w_src[d * 96 + t];
    s_in[t] = s;
    int m = t >> 4, k = t & 15;
    int r = ridxs[b];
    int idx = rni[r * MM + m];
    int msk = rnm[r * MM + m];
    s_in[96 + t] = (msk == 0) ? -1.0f : xa[(size_t)idx * DD + k];
  }
  __syncthreads();
  if (t < 32) {
    float s = gated_b[t];
    for (int d = 0; d < 192; ++d) s += s_in[d] * gated_w[d * 32 + t];
    s_g[t] = s;
  }
  __syncthreads();
  if (t < 16) s_h[t] = s_g[t] * (1.0f / (1.0f + __expf(-s_g[16 + t])));
  __syncthreads();
  if (t < HH) {
    float s = b1[t];
    for (int k = 0; k < DD; ++k) s += s_h[k] * w1[k * HH + t];
    s_h1[t] = fmaxf(s, 0.f);
  }
  __syncthreads();
  if (t < HH) {
    float s = b2[t];
    for (int k = 0; k < HH; ++k) s += s_h1[k] * w2[k * HH + t];
    s_h2[t] = fmaxf(s, 0.f);
  }
  __syncthreads();
  if (t < MM) {
    float s = ab[t];
    for (int k = 0; k < HH; ++k) s += s_h2[k] * aw[k * MM + t];
    out[b * MM + t] = s;
  }
}

extern "C" void kernel_launch(void* const* d_in, const int* in_sizes, int n_in,
                              void* d_out, int out_size, void* d_ws, size_t ws_size,
                              hipStream_t stream) {
  const float* obs     = (const float*)d_in[0];
  const float* obs_all = (const float*)d_in[1];
  const float* ea      = (const float*)d_in[2];
  const int*   ridxs   = (const int*)d_in[3];
  const int*   rni     = (const int*)d_in[4];
  const int*   rnm     = (const int*)d_in[5];
  const float* A       = (const float*)d_in[6];
  const float* PA      = (const float*)d_in[7];
  const float* w_src   = (const float*)d_in[8];
  const float* b_src   = (const float*)d_in[9];
  const float* w_obs   = (const float*)d_in[10];
  const float* b_obs   = (const float*)d_in[11];
  const float* res_w   = (const float*)d_in[12];
  const float* res_b   = (const float*)d_in[13];
  const float* ge_we   = (const float*)d_in[14];
  const float* ge_be   = (const float*)d_in[15];
  const float* ge_wemb = (const float*)d_in[16];
  const float* ge_bemb = (const float*)d_in[17];
  const float* gated_w = (const float*)d_in[18];
  const float* gated_b = (const float*)d_in[19];
  const float* base_w1 = (const float*)d_in[20];
  const float* base_b1 = (const float*)d_in[21];
  const float* base_w2 = (const float*)d_in[22];
  const float* base_b2 = (const float*)d_in[23];
  const float* act_w   = (const float*)d_in[24];
  const float* act_b   = (const float*)d_in[25];
  float* ws  = (float*)d_ws;
  float* out = (float*)d_out;

  k_setup<<<dim3(1), dim3(128), 0, stream>>>(ge_we, ge_be, ge_wemb, ws);
  k_embed<<<dim3(32), dim3(256), 0, stream>>>(obs_all, w_obs, b_obs, ws);
  k_aggregate<<<dim3(512), dim3(256), 0, stream>>>(ea, A, PA, ws);
  k_combine<<<dim3(32), dim3(256), 0, stream>>>(res_w, res_b, ge_wemb, ge_bemb, ws);
  k_head<<<dim3(8), dim3(128), 0, stream>>>(obs, ridxs, rni, rnm, w_src, b_src,
                                            gated_w, gated_b, base_w1, base_b1,
                                            base_w2, base_b2, act_w, act_b, ws, out);
}